// EntityEncoder_28845000360086
// MI455X (gfx1250) — compile-verified
//
#include <hip/hip_runtime.h>
#include <hip/hip_bf16.h>

// EntityEncoder for MI455X (gfx1250, wave32).
// out[e][0..127] = [ent[1], |az|/180, dist_east(az)/180, ent[3], type_emb[type_id][0..123]]
//
// Pure bandwidth-bound gather (no matrix FLOPs -> WMMA inapplicable).
// Strategy: one wave (32 lanes) per 4 entities, taken in turn -> one coalesced 512B float4 row store each.
//   lane 0   : computes the 4 scalar channels
//   lanes1-31: aligned float4 pieces of the 496B embedding row (L2-resident table)
// Cache policy: NT stores for the 256MB output stream + NT loads for the
// read-once entity stream, RT loads for the embedding table so it stays in L2.

typedef float v4f __attribute__((ext_vector_type(4)));

#define EMBD   128   // output channels per entity
#define EMB_T  124   // type-embedding width
#define ENT_C  5     // entity input channels
#define ENT_PER_WAVE 4
#define BLOCK  256   // 8 waves per block (wave32)

__global__ __launch_bounds__(BLOCK)
void EntityEncoder_gfx1250_kernel(const float* __restrict__ entities,
                                  const float* __restrict__ type_emb,
                                  float* __restrict__ out,
                                  int total_entities)
{
    const int lane   = threadIdx.x & 31;
    const int waveId = (blockIdx.x * (BLOCK >> 5)) + (threadIdx.x >> 5);
    const int e0     = waveId * ENT_PER_WAVE;
    if (e0 >= total_entities) return;

    // Prefetch this wave's entity rows (ENT_PER_WAVE * 20B, 1-2 cachelines)
    // while the first dependent loads are being issued. -> global_prefetch_b8
    __builtin_prefetch(entities + (size_t)e0 * ENT_C, 0, 0);

#pragma unroll
    for (int i = 0; i < ENT_PER_WAVE; ++i) {
        const int e = e0 + i;
        if (e >= total_entities) break;

        const float* ent = entities + (size_t)e * ENT_C;
        // Same-address-per-wave loads coalesce into single requests.
        // NT hint: entity stream is read exactly once.
        const float fa = __builtin_nontemporal_load(ent + 1);
        const float az = __builtin_nontemporal_load(ent + 2);
        const float fb = __builtin_nontemporal_load(ent + 3);
        const float tf = __builtin_nontemporal_load(ent + 4);
        const int   t  = (int)tf;  // truncation matches astype(int32)

        // Embedding row: base = type_emb + t*124 floats = t*496B (16B aligned).
        // Lanes 1..31 cover floats 0..123 as aligned float4s. RT policy: keep
        // the 496KB table resident in the 192MB L2.
        const v4f* __restrict__ row =
            (const v4f*)(type_emb + (size_t)t * EMB_T);
        const int ei = (lane == 0) ? 0 : (lane - 1);
        v4f v = row[ei];

        if (lane == 0) {
            const float dn = __builtin_fabsf(az) * (1.0f / 180.0f);
            float de = (az >= -90.0f)
                           ? __builtin_fabsf(90.0f - az)
                           : (90.0f + __builtin_fabsf(az + 180.0f));
            de *= (1.0f / 180.0f);
            v = (v4f){fa, dn, de, fb};
        }

        // One coalesced 512B row store per wave; NT so the 256MB output
        // stream does not evict the embedding table from L2.
        v4f* dst = (v4f*)(out + (size_t)e * EMBD) + lane;
        __builtin_nontemporal_store(v, dst);
    }
}

extern "C" void kernel_launch(void* const* d_in, const int* in_sizes, int n_in,
                              void* d_out, int out_size, void* d_ws, size_t ws_size,
                              hipStream_t stream)
{
    const float* entities = (const float*)d_in[0];   // [B, N, 5] f32
    const float* type_emb = (const float*)d_in[1];   // [1000, 124] f32
    float*       out      = (float*)d_out;           // [B, N, 128] f32

    const int total_entities = in_sizes[0] / ENT_C;  // B*N = 524288
    const int waves  = (total_entities + ENT_PER_WAVE - 1) / ENT_PER_WAVE;
    const int blocks = (waves + (BLOCK / 32) - 1) / (BLOCK / 32);

    EntityEncoder_gfx1250_kernel<<<blocks, BLOCK, 0, stream>>>(
        entities, type_emb, out, total_entities);
}